// Canny_61100204753382
// MI455X (gfx1250) — compile-verified
//
#include <hip/hip_runtime.h>
#include <hip/hip_bf16.h>
#include <math.h>

typedef __attribute__((ext_vector_type(2))) float v2f;
typedef __attribute__((ext_vector_type(8))) float v8f;

#define IMG_H 512
#define IMG_W 512
#define HW    (IMG_H * IMG_W)
#define NB    8
#define NC    3
#define LOW_T  0.1f
#define HIGH_T 0.3f

// NMS neighbor offsets per direction k: (dr, dc) for the -1 tap
__device__ __constant__ int c_dr[8] = {0, 1, 1, 1, 0, -1, -1, -1};
__device__ __constant__ int c_dc[8] = {1, 1, 0, -1, -1, -1, 0, 1};

// Gaussian band value g[d] for d in [0,8] (std=1.0, 9 taps), else 0.
// Branchless select chain over inline constants -> pure VALU, no memory.
__device__ __forceinline__ float bandval(int d) {
    float v = 0.0f;
    v = (d == 0 || d == 8) ? 3.35462628e-4f : v;
    v = (d == 1 || d == 7) ? 1.11089965e-2f : v;
    v = (d == 2 || d == 6) ? 1.35335283e-1f : v;
    v = (d == 3 || d == 5) ? 6.06530660e-1f : v;
    v = (d == 4)           ? 1.0f           : v;
    return v;
}

// ---------------------------------------------------------------------------
// K1: separable 9-tap Gaussian blur via V_WMMA_F32_16X16X4_F32.
// One wave per 16x16 output tile. grid = (32, 32, 24 channel-images).
// ---------------------------------------------------------------------------
__global__ __launch_bounds__(32) void k_blur(const float* __restrict__ img,
                                             float* __restrict__ blur) {
    __shared__ float inT[24][25];  // 24x24 haloed input tile, width padded to 25
    __shared__ float hT[24][18];   // 24x16 horizontally-blurred tile, width 18

    const int tx = blockIdx.x * 16;
    const int ty = blockIdx.y * 16;
    const int z  = blockIdx.z;
    const float* src = img + (size_t)z * HW;
    const int l = threadIdx.x;

    // load haloed tile (zero pad outside image)
    for (int idx = l; idx < 24 * 24; idx += 32) {
        int i = idx / 24, j = idx - i * 24;
        int gr = ty - 4 + i, gc = tx - 4 + j;
        float v = 0.0f;
        if ((unsigned)gr < (unsigned)IMG_H && (unsigned)gc < (unsigned)IMG_W)
            v = src[gr * IMG_W + gc];
        inT[i][j] = v;
    }

    const int m  = l & 15;         // M (A) or N (B/C/D) index for this lane
    const int kh = (l >> 4) * 2;   // lane-half K offset

    // Per-lane banded Gaussian operand values, kept in registers and reused by
    // the horizontal B operand and the vertical A operand (both = g[K - m]).
    float bx[6], by[6];
    #pragma unroll
    for (int kc = 0; kc < 6; ++kc) {
        const int K = kc * 4 + kh;
        bx[kc] = bandval(K - m);
        by[kc] = bandval(K + 1 - m);
    }
    __syncthreads();

    // horizontal pass: OutH(16x16) = In(16x24) * Gh(24x16), two M-tiles (rows 0..15, 8..23)
    for (int t = 0; t < 2; ++t) {
        const int ro = t * 8;
        v8f acc = {0.f, 0.f, 0.f, 0.f, 0.f, 0.f, 0.f, 0.f};
        #pragma unroll
        for (int kc = 0; kc < 6; ++kc) {
            const int K = kc * 4 + kh;
            v2f a, b;
            a.x = inT[ro + m][K];
            a.y = inT[ro + m][K + 1];
            b.x = bx[kc];           // Gh[K][n] = g[K - n] (banded Toeplitz)
            b.y = by[kc];
            acc = __builtin_amdgcn_wmma_f32_16x16x4_f32(
                false, a, false, b, (short)0, acc, false, false);
        }
        // C/D layout: VGPR v -> row v (lanes 0-15) / v+8 (lanes 16-31), col = lane&15
        const int rbase = ro + (l >> 4) * 8;
        #pragma unroll
        for (int v = 0; v < 8; ++v) hT[rbase + v][m] = acc[v];
    }
    __syncthreads();

    // vertical pass: Out(16x16) = Gv(16x24) * H(24x16); Gv[m][K] = g[K - m]
    v8f acc = {0.f, 0.f, 0.f, 0.f, 0.f, 0.f, 0.f, 0.f};
    #pragma unroll
    for (int kc = 0; kc < 6; ++kc) {
        const int K = kc * 4 + kh;
        v2f a, b;
        a.x = bx[kc];
        a.y = by[kc];
        b.x = hT[K][m];
        b.y = hT[K + 1][m];
        acc = __builtin_amdgcn_wmma_f32_16x16x4_f32(
            false, a, false, b, (short)0, acc, false, false);
    }

    float* dst = blur + (size_t)z * HW;
    const int col   = tx + m;
    const int rbase = ty + (l >> 4) * 8;
    #pragma unroll
    for (int v = 0; v < 8; ++v) dst[(rbase + v) * IMG_W + col] = acc[v];
}

// ---------------------------------------------------------------------------
// K2: Sobel per channel, magnitude summed over channels, quantized orientation
// ---------------------------------------------------------------------------
__global__ void k_sobel(const float* __restrict__ blur, float* __restrict__ gm,
                        unsigned char* __restrict__ kdir) {
    const int p = blockIdx.x * blockDim.x + threadIdx.x;
    if (p >= NB * HW) return;
    const int w = p & (IMG_W - 1);
    const int h = (p >> 9) & (IMG_H - 1);
    const int b = p >> 18;
    const bool interior = (unsigned)(h - 1) < (unsigned)(IMG_H - 2) &&
                          (unsigned)(w - 1) < (unsigned)(IMG_W - 2);

    float mag = 0.f, sgx = 0.f, sgy = 0.f;
    #pragma unroll
    for (int c = 0; c < NC; ++c) {
        const float* s = blur + (size_t)(b * NC + c) * HW;
        float a[3][3];
        if (interior) {
            const float* row = s + (h - 1) * IMG_W + (w - 1);
            #pragma unroll
            for (int r = 0; r < 3; ++r)
                #pragma unroll
                for (int q = 0; q < 3; ++q) a[r][q] = row[r * IMG_W + q];
        } else {
            #pragma unroll
            for (int r = -1; r <= 1; ++r)
                #pragma unroll
                for (int q = -1; q <= 1; ++q) {
                    int hh = h + r, ww = w + q;
                    a[r + 1][q + 1] =
                        ((unsigned)hh < (unsigned)IMG_H && (unsigned)ww < (unsigned)IMG_W)
                            ? s[hh * IMG_W + ww] : 0.0f;
                }
        }
        float gx = (a[0][0] - a[0][2]) + 2.f * (a[1][0] - a[1][2]) + (a[2][0] - a[2][2]);
        float gy = (a[0][0] + 2.f * a[0][1] + a[0][2]) -
                   (a[2][0] + 2.f * a[2][1] + a[2][2]);
        mag += sqrtf(gx * gx + gy * gy);
        sgx += gx;
        sgy += gy;
    }
    gm[p] = mag;
    float ang = atan2f(sgy, sgx) * 57.295779513082320877f + 180.0f;
    int q = (int)rintf(ang * (1.0f / 45.0f));  // in [0, 8]
    kdir[p] = (unsigned char)(q & 7);
}

// ---------------------------------------------------------------------------
// K3: non-max suppression, faithfully replicating the flat-gather indexing:
// pos(b,h,w) = gm[k_pos][h][w] - gm[k_pos][h+dr(b)][w+dc(b)]  (offset from batch!)
// ---------------------------------------------------------------------------
__global__ void k_nms(const float* __restrict__ gm,
                      const unsigned char* __restrict__ kdir,
                      float* __restrict__ thin) {
    const int p = blockIdx.x * blockDim.x + threadIdx.x;
    if (p >= NB * HW) return;
    const int w = p & (IMG_W - 1);
    const int h = (p >> 9) & (IMG_H - 1);
    const int b = p >> 18;

    const int kp = kdir[p];
    const int kn = (kp + 4) & 7;
    const int h2 = h + c_dr[b];
    const int w2 = w + c_dc[b];
    const bool inb = ((unsigned)h2 < (unsigned)IMG_H) && ((unsigned)w2 < (unsigned)IMG_W);
    const int off  = h * IMG_W + w;
    const int off2 = h2 * IMG_W + w2;

    float pos = gm[kp * HW + off] - (inb ? gm[kp * HW + off2] : 0.0f);
    float neg = gm[kn * HW + off] - (inb ? gm[kn * HW + off2] : 0.0f);
    thin[p] = (fminf(pos, neg) > 0.0f) ? gm[b * HW + off] : 0.0f;
}

// ---------------------------------------------------------------------------
// K4: hysteresis (single connectivity step) + border zeroing + binarize
// ---------------------------------------------------------------------------
__global__ void k_hyst(const float* __restrict__ thin, float* __restrict__ out) {
    const int p = blockIdx.x * blockDim.x + threadIdx.x;
    if (p >= NB * HW) return;
    const int w = p & (IMG_W - 1);
    const int h = (p >> 9) & (IMG_H - 1);
    const int b = p >> 18;

    const float s = thin[p];
    const bool strong = s > HIGH_T;
    const bool middle = (s >= LOW_T) && (s <= HIGH_T);
    bool conn = false;
    if (middle) {
        // middle pixels on the image border get zeroed anyway; only the
        // interior needs the neighborhood test, so use unchecked loads there.
        const float* t = thin + (size_t)b * HW;
        if ((unsigned)(h - 1) < (unsigned)(IMG_H - 2) &&
            (unsigned)(w - 1) < (unsigned)(IMG_W - 2)) {
            const float* row = t + (h - 1) * IMG_W + (w - 1);
            #pragma unroll
            for (int r = 0; r < 3; ++r)
                #pragma unroll
                for (int q = 0; q < 3; ++q) {
                    if (r == 1 && q == 1) continue;
                    conn = conn || (row[r * IMG_W + q] > HIGH_T);
                }
        } else {
            #pragma unroll
            for (int r = -1; r <= 1; ++r)
                #pragma unroll
                for (int q = -1; q <= 1; ++q) {
                    if (r == 0 && q == 0) continue;
                    int hh = h + r, ww = w + q;
                    if ((unsigned)hh < (unsigned)IMG_H && (unsigned)ww < (unsigned)IMG_W)
                        conn = conn || (t[hh * IMG_W + ww] > HIGH_T);
                }
        }
    }
    float o = (strong || (middle && conn)) ? 1.0f : 0.0f;
    if (h == 0 || h == IMG_H - 1 || w == 0 || w == IMG_W - 1) o = 0.0f;
    out[p] = o;
}

extern "C" void kernel_launch(void* const* d_in, const int* in_sizes, int n_in,
                              void* d_out, int out_size, void* d_ws, size_t ws_size,
                              hipStream_t stream) {
    const float* img = (const float*)d_in[0];
    float* out = (float*)d_out;

    // workspace layout: blur 24*HW*4 B | gm 8*HW*4 B | kdir 8*HW B
    // thin reuses the blur region (blur is dead after k_sobel).
    float* blur = (float*)d_ws;
    float* gm   = blur + (size_t)NB * NC * HW;
    unsigned char* kdir = (unsigned char*)(gm + (size_t)NB * HW);
    float* thin = blur;

    dim3 gridBlur(IMG_W / 16, IMG_H / 16, NB * NC);
    k_blur<<<gridBlur, 32, 0, stream>>>(img, blur);

    const int N = NB * HW;
    const int blocks = (N + 255) / 256;
    k_sobel<<<blocks, 256, 0, stream>>>(blur, gm, kdir);
    k_nms<<<blocks, 256, 0, stream>>>(gm, kdir, thin);
    k_hyst<<<blocks, 256, 0, stream>>>(thin, out);
}